// VMDK_74603581931967
// MI455X (gfx1250) — compile-verified
//
#include <hip/hip_runtime.h>
#include <hip/hip_bf16.h>

// ---------------------------------------------------------------------------
// VMD-K pipeline for MI455X (gfx1250, wave32).
//
// Reference math collapses: per_row >= 0  =>  cumsum nondecreasing  =>
// argmin(cumsum) == 0 always. So only K[0,:] matters; we skip the 128 MB K
// scan entirely. Remaining cost = one pass over Wv (64 MB) ~= 2.7 us at
// 23.3 TB/s. The mat-vec is done with exact-fp32 V_WMMA_F32_16X16X4_F32.
// ---------------------------------------------------------------------------

typedef __attribute__((ext_vector_type(2))) float v2f;
typedef __attribute__((ext_vector_type(8))) float v8f;

#define VMDK_D        4096   // HIDDEN * VMD_K
#define VMDK_IN       4096   // INPUT_SIZE (== per-branch K length)
#define VMDK_HID      512
#define VMDK_KSPLIT   8      // K-dimension split for parallelism
#define VMDK_KSLICE   (VMDK_IN / VMDK_KSPLIT)   // 512
#define VMDK_ROWGRP   (VMDK_D / 16)             // 256 row groups of 16
#define VMDK_WAVES    (VMDK_ROWGRP * VMDK_KSPLIT) // 2048 waves
#define VMDK_WPB      8                          // waves per block (256 thr)

// Stage 1: partial[slice][row] = sum_{k in slice} Wv[row,k] * x[branch(row),k]
// One wave handles 16 consecutive rows x one 512-wide K slice via 128
// V_WMMA_F32_16X16X4_F32 steps. x lives in column 0 of B; results land in
// column 0 of C (lanes 0 and 16, VGPRs 0..7).
__global__ __launch_bounds__(256) void vmdk_matvec_wmma(
    const float* __restrict__ x,      // [8,4096]
    const float* __restrict__ Wv,     // [8,512,4096] row-major => row r at r*4096
    float* __restrict__ partial)      // [KSPLIT, 4096]
{
  const int lane = threadIdx.x & 31;
  const int wid  = blockIdx.x * VMDK_WPB + (threadIdx.x >> 5); // 0..2047
  const int rowGroup = wid / VMDK_KSPLIT;                      // 0..255
  const int slice    = wid % VMDK_KSPLIT;                      // 0..7
  const int row0     = rowGroup * 16;                          // base output row
  const int branch   = row0 >> 9;                              // row0 / 512
  const float* xb    = x + (size_t)branch * VMDK_IN;
  const int kBase    = slice * VMDK_KSLICE;

  // A-fragment (16x4 fp32): lane m in 0..15 -> row m, K={0,1};
  // lane m+16 -> row m, K={2,3}.  (ISA 7.12.2, 32-bit A 16x4 layout)
  const int m        = lane & 15;
  const int kLaneOff = (lane >> 4) * 2;                        // 0 or 2
  const float* arow  = Wv + (size_t)(row0 + m) * VMDK_IN + kBase + kLaneOff;

  // B-fragment (4x16 fp32): v0 = rows K0(lanes0-15)/K1(lanes16-31),
  // v1 = rows K2/K3, N = lane&15. Only N==0 (lanes 0,16) is nonzero.
  const bool bLane = (lane & 15) == 0;
  const int  bK    = lane >> 4;                                // 0 or 1

  v8f c = {};                                                  // accumulator
  #pragma unroll 4
  for (int k = 0; k < VMDK_KSLICE; k += 4) {
    v2f a = *(const v2f*)(arow + k);                           // 8B per lane
    const int kk = kBase + k + bK;
    float b0 = xb[kk];                                         // uniform-ish load
    float b1 = xb[kk + 2];
    v2f b;
    b.x = bLane ? b0 : 0.0f;                                   // v_cndmask, EXEC stays full
    b.y = bLane ? b1 : 0.0f;
    c = __builtin_amdgcn_wmma_f32_16x16x4_f32(
        /*neg_a=*/false, a, /*neg_b=*/false, b,
        /*c_mod=*/(short)0, c, /*reuse_a=*/false, /*reuse_b=*/false);
  }

  // Column 0 of C: lane 0 -> M=0..7 in c[0..7]; lane 16 -> M=8..15.
  if (bLane) {
    float* dst = partial + (size_t)slice * VMDK_D + row0 + (lane >> 4) * 8;
    #pragma unroll
    for (int i = 0; i < 8; ++i) dst[i] = c[i];
  }
}

// Stage 2: deterministic reduction of K-slices, bias+ReLU, feat = (h-K0)*W,
// dot with Wout, sigmoid. Single block of 256 threads (8 waves).
__global__ __launch_bounds__(256) void vmdk_epilogue(
    const float* __restrict__ partial,  // [KSPLIT, 4096]
    const float* __restrict__ bv,       // [8,512] flat == [4096]
    const float* __restrict__ K0,       // K row 0: [4096]
    const float* __restrict__ W,        // [4096]
    const float* __restrict__ Wout,     // [4096]
    const float* __restrict__ bout,     // [1]
    float* __restrict__ out)            // [1]
{
  __shared__ float red[VMDK_WPB];
  float acc = 0.0f;
  for (int d = threadIdx.x; d < VMDK_D; d += 256) {
    float s = 0.0f;
    #pragma unroll
    for (int sl = 0; sl < VMDK_KSPLIT; ++sl) s += partial[(size_t)sl * VMDK_D + d];
    float h = s + bv[d];
    h = h > 0.0f ? h : 0.0f;                 // ReLU
    float feat = (h - K0[d]) * W[d];         // diff[0][d] (argmin row is 0)
    acc += feat * Wout[d];
  }
  // wave32 tree reduce
  #pragma unroll
  for (int off = 16; off > 0; off >>= 1)
    acc += __shfl_down(acc, off, 32);
  if ((threadIdx.x & 31) == 0) red[threadIdx.x >> 5] = acc;
  __syncthreads();
  if (threadIdx.x == 0) {
    float t = 0.0f;
    #pragma unroll
    for (int i = 0; i < VMDK_WPB; ++i) t += red[i];
    t += bout[0];
    out[0] = 1.0f / (1.0f + __expf(-t));     // sigmoid
  }
}

extern "C" void kernel_launch(void* const* d_in, const int* in_sizes, int n_in,
                              void* d_out, int out_size, void* d_ws, size_t ws_size,
                              hipStream_t stream) {
  const float* x    = (const float*)d_in[0];  // [8,4096]
  const float* Wv   = (const float*)d_in[1];  // [8,512,4096]
  const float* bv   = (const float*)d_in[2];  // [8,512]
  const float* K    = (const float*)d_in[3];  // [8192,4096] (only row 0 read)
  const float* W    = (const float*)d_in[4];  // [4096]
  const float* Wout = (const float*)d_in[5];  // [4096]
  const float* bout = (const float*)d_in[6];  // [1]
  float* out = (float*)d_out;

  float* partial = (float*)d_ws;              // KSPLIT*4096*4 = 128 KB scratch

  const int blocks = VMDK_WAVES / VMDK_WPB;   // 256 blocks x 256 threads
  vmdk_matvec_wmma<<<blocks, 256, 0, stream>>>(x, Wv, partial);
  vmdk_epilogue<<<1, 256, 0, stream>>>(partial, bv, K /*row 0*/, W, Wout, bout, out);
}